// ContextQueryAttention_7748121002672
// MI455X (gfx1250) — compile-verified
//
#include <hip/hip_runtime.h>
#include <hip/hip_bf16.h>

typedef float v2f __attribute__((ext_vector_type(2)));
typedef float v8f __attribute__((ext_vector_type(8)));

#define NEG_INF (-1000000000.0f)

static __device__ __forceinline__ v8f wmma_f32x4(v2f a, v2f b, v8f c) {
    // D = A(16x4 f32) x B(4x16 f32) + C(16x16 f32)
    return __builtin_amdgcn_wmma_f32_16x16x4_f32(false, a, false, b, (short)0, c,
                                                 false, false);
}

// ---------------------------------------------------------------------------
// Kernel A: per (batch, 128-row context tile).
//   S = (ctx*w3) @ q^T + cw1 + qw2 + mask ; a = softmax_q(S) ; c2q = a @ q
//   writes out chunks [0:128]=ctx, [128:256]=c2q, [256:384]=ctx*c2q
//   writes rowmax[b,c] = max_q S  (for the q2c softmax over c)
// ---------------------------------------------------------------------------
#define CPITCH 132   // LDS row pitch in floats (mult of 4, conflict-free)
#define SMEM_A_FLOATS (2 * 128 * CPITCH + 128 + 128 + 128)
#define SMEM_A_BYTES  (SMEM_A_FLOATS * 4)

__global__ __launch_bounds__(256) void cqa_main(
    const float* __restrict__ ctx, const float* __restrict__ qry,
    const float* __restrict__ w, const int* __restrict__ qmask,
    float* __restrict__ out, float* __restrict__ rowmax_ws) {
    const int Cn = 2048, Qn = 128, Dn = 128;
    extern __shared__ float smem[];
    float* csh  = smem;                    // 128 x CPITCH  (context, later: probs a)
    float* qsh  = smem + 128 * CPITCH;     // 128 x CPITCH  (query)
    float* cw1s = qsh + 128 * CPITCH;      // 128
    float* qw2m = cw1s + 128;              // 128 (q·w2 + mask bias)
    float* w3s  = qw2m + 128;              // 128

    const int b  = blockIdx.y;
    const int c0 = blockIdx.x * 128;
    const int tid  = threadIdx.x;
    const int wave = tid >> 5;
    const int lane = tid & 31;
    const int lh   = lane & 15;
    const int hi   = lane >> 4;

    // ---- stage context tile and query tile into LDS (b128 loads/stores) ----
    const float* ctxb = ctx + ((size_t)b * Cn + c0) * Dn;
    const float* qryb = qry + (size_t)b * Qn * Dn;
    for (int i = tid; i < 128 * 32; i += 256) {        // 32 float4 per row
        int row = i >> 5, c4 = (i & 31) << 2;
        float4 v = ((const float4*)ctxb)[i];
        *(float4*)&csh[row * CPITCH + c4] = v;
        float4 u = ((const float4*)qryb)[i];
        *(float4*)&qsh[row * CPITCH + c4] = u;
    }
    if (tid < 128) w3s[tid] = w[256 + tid];
    __syncthreads();

    // ---- per-row dot biases ----
    if (tid < 128) {
        float s = 0.f;
        for (int k = 0; k < 128; ++k) s += csh[tid * CPITCH + k] * w[k];
        cw1s[tid] = s;
    } else {
        int q = tid - 128;
        float s = 0.f;
        for (int k = 0; k < 128; ++k) s += qsh[q * CPITCH + k] * w[128 + k];
        s += (1.0f - (float)qmask[b * Qn + q]) * NEG_INF;
        qw2m[q] = s;
    }
    __syncthreads();

    // ---- S GEMM: M=16 (per wave), N=128, K=128, fp32 WMMA 16x16x4 ----
    const int m_base = wave * 16;
    const int arow   = m_base + lh;        // A-frag row for this lane
    v8f acc[8];
    const v8f vzero = {0.f, 0.f, 0.f, 0.f, 0.f, 0.f, 0.f, 0.f};
#pragma unroll
    for (int nt = 0; nt < 8; ++nt) acc[nt] = vzero;

    for (int kb = 0; kb < 32; ++kb) {
        int ka = kb * 4 + hi * 2;          // lanes 0-15: k,k+1 ; 16-31: k+2,k+3
        v2f a;
        a.x = csh[arow * CPITCH + ka]     * w3s[ka];
        a.y = csh[arow * CPITCH + ka + 1] * w3s[ka + 1];
#pragma unroll
        for (int nt = 0; nt < 8; ++nt) {
            int nb = nt * 16 + lh;         // S[m,n] = sum_k A[m,k]*q[n,k]
            v2f bb;
            bb.x = qsh[nb * CPITCH + ka];
            bb.y = qsh[nb * CPITCH + ka + 1];
            acc[nt] = wmma_f32x4(a, bb, acc[nt]);
        }
    }

    // ---- bias add + softmax over q, write probs into csh, rowmax to ws ----
#pragma unroll
    for (int r = 0; r < 8; ++r) {
        int m = m_base + r + hi * 8;       // C/D layout: vgpr r -> rows r / r+8
        float cw1v = cw1s[m];
        float sv[8];
        float lmax = -3.0e38f;
#pragma unroll
        for (int nt = 0; nt < 8; ++nt) {
            sv[nt] = acc[nt][r] + cw1v + qw2m[nt * 16 + lh];
            lmax = fmaxf(lmax, sv[nt]);
        }
#pragma unroll
        for (int off = 1; off < 16; off <<= 1)
            lmax = fmaxf(lmax, __shfl_xor(lmax, off, 32));
        float lsum = 0.f;
#pragma unroll
        for (int nt = 0; nt < 8; ++nt) {
            sv[nt] = __expf(sv[nt] - lmax);
            lsum += sv[nt];
        }
#pragma unroll
        for (int off = 1; off < 16; off <<= 1)
            lsum += __shfl_xor(lsum, off, 32);
        float inv = 1.0f / lsum;
#pragma unroll
        for (int nt = 0; nt < 8; ++nt)
            csh[m * CPITCH + nt * 16 + lh] = sv[nt] * inv;   // own-wave rows only
        if (lh == 0) rowmax_ws[(size_t)b * Cn + c0 + m] = lmax;
    }

    // ---- c2q GEMM: c2q[m,d] = sum_q a[m,q] * qry[q,d] ----
    v8f acc2[8];
#pragma unroll
    for (int nt = 0; nt < 8; ++nt) acc2[nt] = vzero;

    for (int kb = 0; kb < 32; ++kb) {
        int ka = kb * 4 + hi * 2;
        v2f a;
        a.x = csh[arow * CPITCH + ka];
        a.y = csh[arow * CPITCH + ka + 1];
#pragma unroll
        for (int nt = 0; nt < 8; ++nt) {
            int n = nt * 16 + lh;
            v2f bb;
            bb.x = qsh[ka * CPITCH + n];
            bb.y = qsh[(ka + 1) * CPITCH + n];
            acc2[nt] = wmma_f32x4(a, bb, acc2[nt]);
        }
    }

    // ---- write output chunks 0..2 ----
#pragma unroll
    for (int r = 0; r < 8; ++r) {
        int m = m_base + r + hi * 8;
        size_t rowg = (size_t)b * Cn + c0 + m;
        const float* crow = ctx + rowg * Dn;
        float* orow = out + rowg * (4 * Dn);
#pragma unroll
        for (int nt = 0; nt < 8; ++nt) {
            int d = nt * 16 + lh;
            float cv  = crow[d];
            float c2v = acc2[nt][r];
            orow[d]            = cv;
            orow[Dn + d]       = c2v;
            orow[2 * Dn + d]   = cv * c2v;
        }
    }
}

// ---------------------------------------------------------------------------
// Kernel B1: per batch, max & sum(exp) over the 2048 row-maxes
// ---------------------------------------------------------------------------
__global__ __launch_bounds__(256) void cqa_bmax(const float* __restrict__ rowmax,
                                                float* __restrict__ msum) {
    const int b = blockIdx.x, tid = threadIdx.x;
    __shared__ float red[256];
    float lm = -3.0e38f;
    for (int c = tid; c < 2048; c += 256) lm = fmaxf(lm, rowmax[b * 2048 + c]);
    red[tid] = lm;
    __syncthreads();
    for (int s = 128; s > 0; s >>= 1) {
        if (tid < s) red[tid] = fmaxf(red[tid], red[tid + s]);
        __syncthreads();
    }
    float m = red[0];
    __syncthreads();
    float ls = 0.f;
    for (int c = tid; c < 2048; c += 256) ls += __expf(rowmax[b * 2048 + c] - m);
    red[tid] = ls;
    __syncthreads();
    for (int s = 128; s > 0; s >>= 1) {
        if (tid < s) red[tid] += red[tid + s];
        __syncthreads();
    }
    if (tid == 0) { msum[b * 2] = m; msum[b * 2 + 1] = red[0]; }
}

// ---------------------------------------------------------------------------
// Kernel B2: partial q2c over 128-row chunks (deterministic, no atomics)
// ---------------------------------------------------------------------------
__global__ __launch_bounds__(256) void cqa_partial(
    const float* __restrict__ ctx, const float* __restrict__ rowmax,
    const float* __restrict__ msum, float* __restrict__ part) {
    const int t = blockIdx.x, b = blockIdx.y, tid = threadIdx.x;
    __shared__ float wls[128];
    __shared__ float red[256];
    const float m = msum[b * 2];
    const int cbase = t * 128;
    if (tid < 128) wls[tid] = __expf(rowmax[b * 2048 + cbase + tid] - m);
    __syncthreads();
    const int d = tid & 127, half = tid >> 7;
    const float* cb = ctx + ((size_t)b * 2048 + cbase) * 128;
    float acc = 0.f;
    for (int i = half; i < 128; i += 2) acc += wls[i] * cb[(size_t)i * 128 + d];
    red[tid] = acc;
    __syncthreads();
    if (tid < 128)
        part[((size_t)b * 16 + t) * 128 + tid] = red[tid] + red[tid + 128];
}

// ---------------------------------------------------------------------------
// Kernel B3: combine 16 partials -> q2c[b, d]
// ---------------------------------------------------------------------------
__global__ __launch_bounds__(128) void cqa_q2c(const float* __restrict__ part,
                                               const float* __restrict__ msum,
                                               float* __restrict__ q2c) {
    const int b = blockIdx.x, d = threadIdx.x;
    float s = 0.f;
    for (int t = 0; t < 16; ++t) s += part[((size_t)b * 16 + t) * 128 + d];
    q2c[b * 128 + d] = s / msum[b * 2 + 1];
}

// ---------------------------------------------------------------------------
// Kernel C: out chunk 3 = ctx * broadcast(q2c)
// ---------------------------------------------------------------------------
__global__ __launch_bounds__(256) void cqa_chunk3(const float* __restrict__ ctx,
                                                  const float* __restrict__ q2c,
                                                  float* __restrict__ out) {
    size_t idx = (size_t)blockIdx.x * 256 + threadIdx.x;   // 32*2048*128 elems
    int b   = (int)(idx >> 18);
    int rem = (int)(idx & ((1u << 18) - 1));
    int c = rem >> 7;
    int d = rem & 127;
    out[(((size_t)b * 2048 + c) << 9) + 384 + d] = ctx[idx] * q2c[b * 128 + d];
}

extern "C" void kernel_launch(void* const* d_in, const int* in_sizes, int n_in,
                              void* d_out, int out_size, void* d_ws, size_t ws_size,
                              hipStream_t stream) {
    (void)in_sizes; (void)n_in; (void)out_size; (void)ws_size;
    const float* ctx   = (const float*)d_in[0];   // (32,2048,128) f32
    const float* qry   = (const float*)d_in[1];   // (32,128,128)  f32
    const float* w     = (const float*)d_in[2];   // (384,)        f32
    const int*   qmask = (const int*)d_in[3];     // (32,128)      i32
    float* out = (float*)d_out;                   // (32,2048,512) f32

    char* ws = (char*)d_ws;
    float* rowmax = (float*)(ws);                 // 32*2048 f32 = 256 KB
    float* part   = (float*)(ws + 262144);        // 32*16*128 f32 = 256 KB
    float* msum   = (float*)(ws + 524288);        // 32*2 f32
    float* q2c    = (float*)(ws + 524800);        // 32*128 f32

    hipFuncSetAttribute((const void*)cqa_main,
                        hipFuncAttributeMaxDynamicSharedMemorySize, SMEM_A_BYTES);

    cqa_main<<<dim3(16, 32), 256, SMEM_A_BYTES, stream>>>(ctx, qry, w, qmask, out,
                                                          rowmax);
    cqa_bmax<<<32, 256, 0, stream>>>(rowmax, msum);
    cqa_partial<<<dim3(16, 32), 256, 0, stream>>>(ctx, rowmax, msum, part);
    cqa_q2c<<<32, 128, 0, stream>>>(part, msum, q2c);
    cqa_chunk3<<<32768, 256, 0, stream>>>(ctx, q2c, out);
}